// UpFirDn2D_63127429317258
// MI455X (gfx1250) — compile-verified
//
#include <hip/hip_runtime.h>
#include <stdint.h>

// ---------------------------------------------------------------------------
// upfirdn2d (up=2, 4x4 FIR, dn=1) for MI455X / gfx1250.
//
// Memory-bound problem: 335 MB moved vs ~0.54 GFLOP => ~14.4 us floor at
// 23.3 TB/s.  Strategy: async-copy input strips into LDS (gfx1250
// GLOBAL_LOAD_ASYNC_TO_LDS_B128, ASYNCcnt), polyphase 2x2 stencil from LDS,
// stream output with nontemporal 128-bit stores.
// ---------------------------------------------------------------------------

typedef __attribute__((ext_vector_type(4))) float v4f;

#define BATCH 16
#define CH    64
#define H     128
#define W     128
#define OH    256
#define OW    256

#define ROWS_PER_BLOCK 8
#define TILE_ROWS      (ROWS_PER_BLOCK + 2)  // 10 (one halo row top+bottom)
#define TILE_STRIDE    132                   // floats; 528 B, keeps 16B align

// Raw 32-bit LDS byte offset of a pointer into a __shared__ array
// (addrspacecast generic->local; low 32 bits of the flat LDS aperture
// address ARE the LDS offset per CDNA5 ISA 10.2).
__device__ __forceinline__ uint32_t lds_off(const void* p) {
    return (uint32_t)(uintptr_t)(__attribute__((address_space(3))) const void*)p;
}

// One 16-byte async global->LDS transfer per enabled lane (GVS addressing:
// 64-bit SGPR base + 32-bit per-lane byte offset). Tracked by ASYNCcnt.
__device__ __forceinline__ void async_load_b128(uint32_t lds_byte_off,
                                                const float* sbase,
                                                int voffset_bytes) {
    asm volatile("global_load_async_to_lds_b128 %0, %1, %2"
                 :
                 : "v"(lds_byte_off), "v"(voffset_bytes), "s"(sbase)
                 : "memory");
}

__device__ __forceinline__ void wait_asynccnt0() {
    asm volatile("s_wait_asynccnt 0" ::: "memory");
}

extern "C" __global__ void __launch_bounds__(256)
upfirdn2d_up2_f4_kernel(const float* __restrict__ x,
                        const float* __restrict__ filt,
                        float* __restrict__ out) {
    __shared__ alignas(16) float tile[TILE_ROWS * TILE_STRIDE];

    const int tid   = threadIdx.x;
    const int lane  = tid & 31;   // 0..31 : 4 input cols each
    const int wave  = tid >> 5;   // 0..7  : one input row each
    const int strip = blockIdx.x; // 0..15 : 8-row strip of the plane
    const int plane = blockIdx.y; // 0..1023 = b*CH + c
    const int c     = plane & (CH - 1);
    const int r0    = strip * ROWS_PER_BLOCK;

    const float* xplane = x + (size_t)plane * (H * W);

    // ---- stage rows [r0-1 .. r0+8] into LDS tile rows [0..9] -------------
    // Waves 0..7 handle tile rows 0..7; waves 0..1 also handle rows 8..9.
    #pragma unroll
    for (int pass = 0; pass < 2; ++pass) {
        const int tr = wave + pass * 8;       // tile row
        if (tr < TILE_ROWS) {
            const int gr = r0 - 1 + tr;       // global input row (may be OOB)
            float* dstp  = &tile[tr * TILE_STRIDE + lane * 4];
            if (gr >= 0 && gr < H) {
                async_load_b128(lds_off(dstp), xplane, (gr * W + lane * 4) * 4);
            } else {
                *(v4f*)dstp = (v4f)0.0f;      // zero halo row (top/bottom pad)
            }
        }
    }
    wait_asynccnt0();     // drain this wave's ASYNCcnt before the barrier
    __syncthreads();

    // ---- per-channel 4x4 filter (uniform per block, L2-resident) ---------
    const float* fp = filt + c * 16;
    float f[16];
    #pragma unroll
    for (int k = 0; k < 16; ++k) f[k] = fp[k];

    // ---- polyphase 2x2 stencil: 4 input cols -> 2x8 output block ---------
    const int i  = r0 + wave;   // input row covered by this thread
    const int tr = wave + 1;    // its tile row
    const int j0 = lane * 4;    // first input col covered

    // a[r][k] = x(i-1+r, j0-1+k), k = 0..5, zero outside [0,W)
    float a[3][6];
    #pragma unroll
    for (int r = 0; r < 3; ++r) {
        const float* row = &tile[(tr - 1 + r) * TILE_STRIDE];
        const v4f m = *(const v4f*)&row[j0];
        a[r][1] = m.x; a[r][2] = m.y; a[r][3] = m.z; a[r][4] = m.w;
        const int   cl = (j0 > 0) ? j0 - 1 : 0;        // clamp addr, mask value
        const int   cr = (lane < 31) ? j0 + 4 : W - 1;
        const float vl = row[cl];
        const float vr = row[cr];
        a[r][0] = (j0 > 0)    ? vl : 0.0f;
        a[r][5] = (lane < 31) ? vr : 0.0f;
    }

    alignas(16) float otop[8];  // output row 2*i,   cols 2*j0 .. 2*j0+7
    alignas(16) float obot[8];  // output row 2*i+1
    #pragma unroll
    for (int q = 0; q < 4; ++q) {
        const float um1_l = a[0][q], um1_c = a[0][q + 1], um1_r = a[0][q + 2];
        const float u0_l  = a[1][q], u0_c  = a[1][q + 1], u0_r  = a[1][q + 2];
        const float up1_l = a[2][q], up1_c = a[2][q + 1], up1_r = a[2][q + 2];
        // out(2i+0, 2j+0): taps (ky,kx) in {0,2}x{0,2}
        otop[2 * q + 0] = f[0]  * um1_l + f[2]  * um1_c
                        + f[8]  * u0_l  + f[10] * u0_c;
        // out(2i+0, 2j+1): {0,2}x{1,3}
        otop[2 * q + 1] = f[1]  * um1_c + f[3]  * um1_r
                        + f[9]  * u0_c  + f[11] * u0_r;
        // out(2i+1, 2j+0): {1,3}x{0,2}
        obot[2 * q + 0] = f[4]  * u0_l  + f[6]  * u0_c
                        + f[12] * up1_l + f[14] * up1_c;
        // out(2i+1, 2j+1): {1,3}x{1,3}
        obot[2 * q + 1] = f[5]  * u0_c  + f[7]  * u0_r
                        + f[13] * up1_c + f[15] * up1_r;
    }

    // ---- streaming, coalesced NT stores (output never re-read) -----------
    float* orow0 = out + (size_t)plane * (OH * OW) + (size_t)(2 * i) * OW + 2 * j0;
    float* orow1 = orow0 + OW;
    __builtin_nontemporal_store(*(const v4f*)&otop[0], (v4f*)orow0);
    __builtin_nontemporal_store(*(const v4f*)&otop[4], (v4f*)(orow0 + 4));
    __builtin_nontemporal_store(*(const v4f*)&obot[0], (v4f*)orow1);
    __builtin_nontemporal_store(*(const v4f*)&obot[4], (v4f*)(orow1 + 4));
}

extern "C" void kernel_launch(void* const* d_in, const int* in_sizes, int n_in,
                              void* d_out, int out_size, void* d_ws, size_t ws_size,
                              hipStream_t stream) {
    (void)in_sizes; (void)n_in; (void)out_size; (void)d_ws; (void)ws_size;
    const float* x    = (const float*)d_in[0];
    const float* filt = (const float*)d_in[1];
    float*       out  = (float*)d_out;

    dim3 grid(H / ROWS_PER_BLOCK, BATCH * CH);  // (16, 1024)
    upfirdn2d_up2_f4_kernel<<<grid, 256, 0, stream>>>(x, filt, out);
}